// LGNLayer_9594956939813
// MI455X (gfx1250) — compile-verified
//
#include <hip/hip_runtime.h>

#define N_RET 8192
#define N_LGN 2048
#define ETA   0.1f
#define MU_WTS 2.5f

typedef __attribute__((ext_vector_type(2))) float v2f;
typedef __attribute__((ext_vector_type(8))) float v8f;

// One wave computes partial dot products of 16 consecutive rows of W against v
// over K range [k0, k1) using V_WMMA_F32_16X16X4_F32.
//   A (16x4): row 0 = v[k..k+3], other rows 0.  Lane layout: m=lane&15,
//             k = 2*(lane>>4) + vgpr  -> only lanes 0 and 16 carry data.
//   B (4x16): B[k][n] = W[rowBase+n][kcol+k].  Lane layout: n=lane&15,
//             k = 2*(lane>>4) + vgpr  -> each lane loads a b64 from its row.
//   D[0][n] (VGPR0, lanes 0..15) accumulates the 16 row partials.
__device__ __forceinline__ float wmma_rowdots(const float* __restrict__ W,
                                              const float* __restrict__ v,
                                              int rowBase, int k0, int k1,
                                              int cols, int lane) {
  const int half = lane >> 4;     // 0 or 1: selects K pair {0,1} vs {2,3}
  const int l15  = lane & 15;
  const float* bptr = W + (size_t)(rowBase + l15) * (size_t)cols + k0 + 2 * half;
  const float* aptr = v + k0 + 2 * half;
  const bool isRow0 = (l15 == 0);
  const v2f zz = (v2f){0.0f, 0.0f};

  v8f acc0 = {}; v8f acc1 = {}; v8f acc2 = {}; v8f acc3 = {};
  for (int k = k0; k < k1; k += 16) {
    v2f b0 = *(const v2f*)(bptr);
    v2f b1 = *(const v2f*)(bptr + 4);
    v2f b2 = *(const v2f*)(bptr + 8);
    v2f b3 = *(const v2f*)(bptr + 12);
    v2f a0 = *(const v2f*)(aptr);
    v2f a1 = *(const v2f*)(aptr + 4);
    v2f a2 = *(const v2f*)(aptr + 8);
    v2f a3 = *(const v2f*)(aptr + 12);
    a0 = isRow0 ? a0 : zz;   // keep EXEC full; mask via cndmask, not branches
    a1 = isRow0 ? a1 : zz;
    a2 = isRow0 ? a2 : zz;
    a3 = isRow0 ? a3 : zz;
    acc0 = __builtin_amdgcn_wmma_f32_16x16x4_f32(false, a0, false, b0, (short)0, acc0, false, false);
    acc1 = __builtin_amdgcn_wmma_f32_16x16x4_f32(false, a1, false, b1, (short)0, acc1, false, false);
    acc2 = __builtin_amdgcn_wmma_f32_16x16x4_f32(false, a2, false, b2, (short)0, acc2, false, false);
    acc3 = __builtin_amdgcn_wmma_f32_16x16x4_f32(false, a3, false, b3, (short)0, acc3, false, false);
    bptr += 16;
    aptr += 16;
  }
  v8f s = (acc0 + acc1) + (acc2 + acc3);
  return s[0];   // lanes 0..15 hold rows rowBase..rowBase+15
}

// Kernel 1: new_firing = (retina_weights @ is_firing > retina_threshold)
// 512 blocks x 256 threads (8 waves); each block owns 16 rows, waves K-split.
__global__ __launch_bounds__(256) void retina_matvec_kernel(
    const float* __restrict__ W, const float* __restrict__ firing_in,
    const float* __restrict__ thr, float* __restrict__ firing_out) {
  __shared__ float lds[8 * 16];
  const int tid  = threadIdx.x;
  const int lane = tid & 31;
  const int wave = tid >> 5;
  const int rowBase = blockIdx.x * 16;
  const int kspan = N_RET / 8;                 // 1024 per wave
  const int k0 = wave * kspan;
  float r = wmma_rowdots(W, firing_in, rowBase, k0, k0 + kspan, N_RET, lane);
  if (lane < 16) lds[wave * 16 + lane] = r;
  __syncthreads();
  if (tid < 16) {
    float x = 0.0f;
    #pragma unroll
    for (int w = 0; w < 8; ++w) x += lds[w * 16 + tid];
    const int row = rowBase + tid;
    firing_out[row] = (x > thr[row]) ? 1.0f : 0.0f;
  }
}

// Kernel 2: lgn_act = relu(lgn_weights @ new_firing)
__global__ __launch_bounds__(256) void lgn_matvec_kernel(
    const float* __restrict__ W, const float* __restrict__ firing,
    float* __restrict__ lgn_act_out) {
  __shared__ float lds[8 * 16];
  const int tid  = threadIdx.x;
  const int lane = tid & 31;
  const int wave = tid >> 5;
  const int rowBase = blockIdx.x * 16;
  const int kspan = N_RET / 8;
  const int k0 = wave * kspan;
  float r = wmma_rowdots(W, firing, rowBase, k0, k0 + kspan, N_RET, lane);
  if (lane < 16) lds[wave * 16 + lane] = r;
  __syncthreads();
  if (tid < 16) {
    float x = 0.0f;
    #pragma unroll
    for (int w = 0; w < 8; ++w) x += lds[w * 16 + tid];
    lgn_act_out[rowBase + tid] = fmaxf(x, 0.0f);
  }
}

// Kernel 3 (single block): argmax of relu(lgn_act - thr) with first-index
// tie-break, winner-row mean -> scale, and lgn_threshold_new output.
__global__ __launch_bounds__(256) void argmax_stats_kernel(
    const float* __restrict__ lgn_act, const float* __restrict__ lgn_thr,
    const float* __restrict__ lgnW, const float* __restrict__ firing,
    float* __restrict__ out_thr, int* __restrict__ wsI, float* __restrict__ wsF) {
  __shared__ float sv[256];
  __shared__ int   si[256];
  const int tid = threadIdx.x;

  float best = -1.0f;            // act >= 0, so every real value wins once
  int   bidx = 0x7fffffff;
  for (int i = tid; i < N_LGN; i += 256) {   // increasing i: '>' keeps first max
    float a = fmaxf(lgn_act[i] - lgn_thr[i], 0.0f);
    if (a > best) { best = a; bidx = i; }
  }
  sv[tid] = best; si[tid] = bidx;
  __syncthreads();
  for (int s = 128; s > 0; s >>= 1) {
    if (tid < s) {
      float ov = sv[tid + s]; int oi = si[tid + s];
      if (ov > sv[tid] || (ov == sv[tid] && oi < si[tid])) { sv[tid] = ov; si[tid] = oi; }
    }
    __syncthreads();
  }
  const int   idx = si[0];
  const float val = sv[0];
  __syncthreads();

  // mean of the updated winner row (two half-additions, as in the reference)
  float partial = 0.0f;
  const float* row = lgnW + (size_t)idx * N_RET;
  for (int c = tid; c < N_RET; c += 256) {
    float h = 0.5f * (ETA * val * firing[c]);
    partial += (row[c] + h) + h;
  }
  sv[tid] = partial;
  __syncthreads();
  for (int s = 128; s > 0; s >>= 1) {
    if (tid < s) sv[tid] += sv[tid + s];
    __syncthreads();
  }
  if (tid == 0) {
    wsI[0] = idx;
    wsF[1] = val;
    wsF[2] = MU_WTS * (float)N_RET / sv[0];   // MU / mean(row')
  }

  // lgn_threshold_new
  for (int i = tid; i < N_LGN; i += 256) {
    float t = lgn_thr[i];
    if (val > 0.0f && i == idx) t += 0.005f * val;
    out_thr[i] = t;
  }
}

// Kernel 4: stream lgn_weights -> out with winner-row update. float4 I/O;
// the 64 MB source was just touched by kernels 2/3, so it rides in L2.
__global__ __launch_bounds__(256) void weights_out_kernel(
    const float* __restrict__ W, const float* __restrict__ firing,
    const int* __restrict__ wsI, const float* __restrict__ wsF,
    float* __restrict__ out) {
  const size_t e4 = (size_t)blockIdx.x * 256 + threadIdx.x;
  const size_t n4 = (size_t)N_LGN * (N_RET / 4);
  if (e4 >= n4) return;
  const int r = (int)(e4 >> 11);               // N_RET/4 = 2048 float4 per row
  float4 w = ((const float4*)W)[e4];
  const int   idx = wsI[0];
  const float val = wsF[1];
  if (val > 0.0f && r == idx) {
    const float scale = wsF[2];
    const int c4 = (int)(e4 & 2047);
    const float4 f = ((const float4*)firing)[c4];
    const float hv = ETA * val;
    w.x = (w.x + 0.5f * hv * f.x + 0.5f * hv * f.x) * scale;
    w.y = (w.y + 0.5f * hv * f.y + 0.5f * hv * f.y) * scale;
    w.z = (w.z + 0.5f * hv * f.z + 0.5f * hv * f.z) * scale;
    w.w = (w.w + 0.5f * hv * f.w + 0.5f * hv * f.w) * scale;
  }
  ((float4*)out)[e4] = w;
}

extern "C" void kernel_launch(void* const* d_in, const int* in_sizes, int n_in,
                              void* d_out, int out_size, void* d_ws, size_t ws_size,
                              hipStream_t stream) {
  const float* retW    = (const float*)d_in[0];   // [8192, 8192]
  const float* retThr  = (const float*)d_in[1];   // [8192]
  const float* lgnW    = (const float*)d_in[2];   // [2048, 8192]
  const float* lgnThr  = (const float*)d_in[3];   // [2048]
  const float* firing0 = (const float*)d_in[4];   // [8192]

  float* out         = (float*)d_out;
  float* out_firing  = out;                                        // [8192]
  float* out_lgn_act = out + N_RET;                                // [2048]
  float* out_weights = out + N_RET + N_LGN;                        // [2048*8192]
  float* out_thr     = out + N_RET + N_LGN + (size_t)N_LGN * N_RET;// [2048]

  int*   wsI = (int*)d_ws;
  float* wsF = (float*)d_ws;

  retina_matvec_kernel<<<N_RET / 16, 256, 0, stream>>>(retW, firing0, retThr, out_firing);
  lgn_matvec_kernel<<<N_LGN / 16, 256, 0, stream>>>(lgnW, out_firing, out_lgn_act);
  argmax_stats_kernel<<<1, 256, 0, stream>>>(out_lgn_act, lgnThr, lgnW, out_firing,
                                             out_thr, wsI, wsF);
  const int n4 = N_LGN * (N_RET / 4);
  weights_out_kernel<<<(n4 + 255) / 256, 256, 0, stream>>>(lgnW, out_firing, wsI, wsF,
                                                           out_weights);
}